// MyListNetLoss_62208306315847
// MI455X (gfx1250) — compile-verified
//
#include <hip/hip_runtime.h>
#include <math.h>

// ---------------------------------------------------------------------------
// ListNet loss, batch=1, N=16M:
//   loss = T*log(S2) - dot,  where p = softmax(x), S2 = sum(exp(p)),
//   dot = sum(g*p), T = sum(g).
// Memory-bound streaming reduction: 128MB compulsory HBM traffic (~5.5us at
// 23.3 TB/s). x (64MB) fits in the 192MB L2, so pass 2 re-reads x from L2;
// g is loaded non-temporally to avoid evicting x. Final partial reduction is
// done with v_wmma_f32_16x16x4_f32 (ones-matrix row-sum trick).
// ---------------------------------------------------------------------------

typedef __attribute__((ext_vector_type(4))) float f4;
typedef __attribute__((ext_vector_type(2))) float v2f;
typedef __attribute__((ext_vector_type(8))) float v8f;

#define NB 2048      // blocks for the streaming kernels (also # partials)
#define BT 256       // threads per block (8 wave32s)
#define NW (BT / 32)

#define NEG_BIG (-3.0e38f)

// ws layout (floats):
// [0,NB)            : block max     (k1)
// [NB,2NB)          : block sumexp  (k1)
// [2NB]             : M
// [2NB+1]           : S1
// [2NB+2, +NB)      : s2 partials   (k3)
// [+NB, +2NB)       : dot partials
// [+2NB, +3NB)      : t partials

__device__ __forceinline__ void online_combine(float& m, float& s, float om, float os) {
  float nm = fmaxf(m, om);
  s = s * __expf(m - nm) + os * __expf(om - nm);
  m = nm;
}

// ---------------- pass 1: online max + sum(exp(x - max)) -------------------
__global__ void k1_maxsum(const float* __restrict__ x, float* __restrict__ ws, int n) {
  const int n4 = n >> 2;
  const f4* x4 = (const f4*)x;
  float m = NEG_BIG, s = 0.0f;
  const int stride = gridDim.x * blockDim.x;
  for (int i = blockIdx.x * blockDim.x + threadIdx.x; i < n4; i += stride) {
    f4 v = x4[i];
    float lm = fmaxf(fmaxf(v.x, v.y), fmaxf(v.z, v.w));
    float nm = fmaxf(m, lm);
    float e = __expf(v.x - nm) + __expf(v.y - nm) + __expf(v.z - nm) + __expf(v.w - nm);
    s = s * __expf(m - nm) + e;
    m = nm;
  }
  // scalar tail (N is divisible by 4 here, but stay generic)
  for (int i = (n4 << 2) + blockIdx.x * blockDim.x + threadIdx.x; i < n; i += stride) {
    float v = x[i];
    float nm = fmaxf(m, v);
    s = s * __expf(m - nm) + __expf(v - nm);
    m = nm;
  }
  // wave32 butterfly
  for (int off = 16; off; off >>= 1) {
    float om = __shfl_xor(m, off, 32);
    float os = __shfl_xor(s, off, 32);
    online_combine(m, s, om, os);
  }
  __shared__ float sm[NW], ss[NW];
  const int wid = threadIdx.x >> 5, lid = threadIdx.x & 31;
  if (lid == 0) { sm[wid] = m; ss[wid] = s; }
  __syncthreads();
  if (threadIdx.x == 0) {
    float bm = sm[0], bs = ss[0];
    for (int w = 1; w < NW; ++w) online_combine(bm, bs, sm[w], ss[w]);
    ws[blockIdx.x]      = bm;
    ws[NB + blockIdx.x] = bs;
  }
}

// ---------------- pass 1b: combine block partials -> M, S1 -----------------
__global__ void k2_combine(float* __restrict__ ws) {
  const int l = threadIdx.x;  // 32 threads
  float m = NEG_BIG, s = 0.0f;
  for (int b = l; b < NB; b += 32) online_combine(m, s, ws[b], ws[NB + b]);
  for (int off = 16; off; off >>= 1) {
    float om = __shfl_xor(m, off, 32);
    float os = __shfl_xor(s, off, 32);
    online_combine(m, s, om, os);
  }
  if (l == 0) { ws[2 * NB] = m; ws[2 * NB + 1] = s; }
}

// ---------------- pass 2: S2, dot, T partials ------------------------------
__global__ void k3_main(const float* __restrict__ x, const float* __restrict__ g,
                        float* __restrict__ ws, int n) {
  const float M     = ws[2 * NB];
  const float invS1 = 1.0f / ws[2 * NB + 1];
  const int n4 = n >> 2;
  const f4* x4 = (const f4*)x;
  const f4* g4 = (const f4*)g;
  float s2 = 0.0f, dot = 0.0f, t = 0.0f;
  const int stride = gridDim.x * blockDim.x;
  for (int i = blockIdx.x * blockDim.x + threadIdx.x; i < n4; i += stride) {
    f4 xv = x4[i];                                   // RT: keep x hot in L2
    f4 gv = __builtin_nontemporal_load(&g4[i]);      // NT: g is read exactly once
    float p0 = __expf(xv.x - M) * invS1;
    float p1 = __expf(xv.y - M) * invS1;
    float p2 = __expf(xv.z - M) * invS1;
    float p3 = __expf(xv.w - M) * invS1;
    s2  += __expf(p0) + __expf(p1) + __expf(p2) + __expf(p3);
    dot += gv.x * p0 + gv.y * p1 + gv.z * p2 + gv.w * p3;
    t   += gv.x + gv.y + gv.z + gv.w;
  }
  for (int i = (n4 << 2) + blockIdx.x * blockDim.x + threadIdx.x; i < n; i += stride) {
    float p = __expf(x[i] - M) * invS1;
    float gv = g[i];
    s2 += __expf(p); dot += gv * p; t += gv;
  }
  for (int off = 16; off; off >>= 1) {
    s2  += __shfl_xor(s2, off, 32);
    dot += __shfl_xor(dot, off, 32);
    t   += __shfl_xor(t, off, 32);
  }
  __shared__ float a0[NW], a1[NW], a2[NW];
  const int wid = threadIdx.x >> 5, lid = threadIdx.x & 31;
  if (lid == 0) { a0[wid] = s2; a1[wid] = dot; a2[wid] = t; }
  __syncthreads();
  if (threadIdx.x == 0) {
    float b0 = a0[0], b1 = a1[0], b2 = a2[0];
    for (int w = 1; w < NW; ++w) { b0 += a0[w]; b1 += a1[w]; b2 += a2[w]; }
    float* base = ws + 2 * NB + 2;
    base[blockIdx.x]          = b0;
    base[NB + blockIdx.x]     = b1;
    base[2 * NB + blockIdx.x] = b2;
  }
}

// ---------------- pass 3: WMMA partial reduction + finalize ----------------
// Sum NB floats with v_wmma_f32_16x16x4_f32 and B = all-ones:
// each WMMA folds 64 values into 16 accumulator rows (D[m][n] = sum_k A[m][k]
// + C[m][n]); summation is permutation-invariant so any lane->A mapping works.
// C/D layout: lane 0 VGPRs 0..7 hold D[0..7][0]; lane 16 holds D[8..15][0].
__device__ __forceinline__ float wmma_sum_nb(const float* __restrict__ p, int lane) {
  v8f c = {};
  v2f b; b[0] = 1.0f; b[1] = 1.0f;
  for (int i = 0; i < NB; i += 64) {
    v2f a;
    const int idx = i + lane * 2;
    a[0] = p[idx];
    a[1] = p[idx + 1];
    c = __builtin_amdgcn_wmma_f32_16x16x4_f32(false, a, false, b,
                                              (short)0, c, false, false);
  }
  float local = c[0] + c[1] + c[2] + c[3] + c[4] + c[5] + c[6] + c[7];
  return __shfl(local, 0, 32) + __shfl(local, 16, 32);
}

__global__ void k4_final(const float* __restrict__ ws, float* __restrict__ out) {
  const int lane = threadIdx.x;  // one full wave32, EXEC all ones
  const float* base = ws + 2 * NB + 2;
  float S2  = wmma_sum_nb(base, lane);
  float dot = wmma_sum_nb(base + NB, lane);
  float T   = wmma_sum_nb(base + 2 * NB, lane);
  if (lane == 0) out[0] = T * logf(S2) - dot;
}

// ---------------------------------------------------------------------------
extern "C" void kernel_launch(void* const* d_in, const int* in_sizes, int n_in,
                              void* d_out, int out_size, void* d_ws, size_t ws_size,
                              hipStream_t stream) {
  const float* x = (const float*)d_in[0];   // predictions [1,N] fp32
  const float* g = (const float*)d_in[1];   // ground_truth_probs [1,N] fp32
  float* ws  = (float*)d_ws;
  float* out = (float*)d_out;
  const int n = in_sizes[0];

  k1_maxsum<<<NB, BT, 0, stream>>>(x, ws, n);
  k2_combine<<<1, 32, 0, stream>>>(ws);
  k3_main<<<NB, BT, 0, stream>>>(x, g, ws, n);
  k4_final<<<1, 32, 0, stream>>>(ws, out);
}